// RNNJANSEN_3341484556376
// MI455X (gfx1250) — compile-verified
//
#include <hip/hip_runtime.h>
#include <math.h>

// ---------------------------------------------------------------------------
// Jansen-Rit neural-mass RNN for MI455X (gfx1250, wave32).
// Persistent cooperative kernel: 16 WGs x 512 threads, one grid barrier per
// time step; all coupling state L2-resident; EEG projection via
// v_wmma_f32_16x16x4_f32 (branchless operand construction, EXEC all-1s).
// ---------------------------------------------------------------------------

#define Nn    512
#define Hh    20
#define Bb    40
#define Ll    500
#define TS    (Hh*Bb)        // 800
#define EEGc  64
#define NWG   16
#define CHUNK 32             // Nn / NWG
#define RP    512            // ring period (pow2 >= Ll)
#define DTc   0.05f

// workspace layout (units: floats)
#define OFF_RSS   64
#define OFF_RSUM  576
#define OFF_INV   1088
#define OFF_DG    1152
#define OFF_WN    2048
#define OFF_RING  (OFF_WN   + Nn*Nn)       // 264192
#define OFF_DEL   (OFF_RING + Nn*RP)       // 526336 (u16 region: Nn*Nn u16)
#define OFF_NOISE (OFF_DEL  + (Nn*Nn)/2)   // 657408
#define OFF_LMT   (OFF_NOISE + TS*Nn)      // 1067008
#define OFF_PART  (OFF_LMT  + EEGc*Nn)     // 1099776 (2 * NWG * Nn, dbl-buffer)

typedef __attribute__((ext_vector_type(2))) float v2f;
typedef __attribute__((ext_vector_type(4))) float v4f;
typedef __attribute__((ext_vector_type(8))) float v8f;

__device__ __forceinline__ float reluf(float x) { return fmaxf(x, 0.0f); }

// ---------------------------------------------------------------------------
__global__ void k0_init(unsigned* bar) {
  if (threadIdx.x < 2) bar[threadIdx.x] = 0u;
}

// per-row: ws = log1p(0.5*(w + w^T)), row sum-of-squares and row sum
__global__ void k1_rows(const float* __restrict__ wbb,
                        const float* __restrict__ sc,
                        float* __restrict__ ws) {
  int i = blockIdx.x;
  int t = threadIdx.x;
  float* WN = ws + OFF_WN;
  float ss = 0.0f, s = 0.0f;
  for (int j = t; j < Nn; j += 256) {
    float wij = expf(wbb[i*Nn + j]) * sc[i*Nn + j];
    float wji = expf(wbb[j*Nn + i]) * sc[j*Nn + i];
    float v = log1pf(0.5f * (wij + wji));   // exactly symmetric in (i,j)
    WN[i*Nn + j] = v;
    ss += v * v;
    s  += v;
  }
  __shared__ float sh1[256], sh2[256];
  sh1[t] = ss; sh2[t] = s;
  __syncthreads();
  for (int k = 128; k > 0; k >>= 1) {      // deterministic tree
    if (t < k) { sh1[t] += sh1[t + k]; sh2[t] += sh2[t + k]; }
    __syncthreads();
  }
  if (t == 0) { ws[OFF_RSS + i] = sh1[0]; ws[OFF_RSUM + i] = sh2[0]; }
}

// frobenius norm + dg_diag
__global__ void k2_norm(float* ws) {
  int t = threadIdx.x;                     // 512 threads
  __shared__ float sh[Nn];
  sh[t] = ws[OFF_RSS + t];
  __syncthreads();
  for (int k = 256; k > 0; k >>= 1) {
    if (t < k) sh[t] += sh[t + k];
    __syncthreads();
  }
  float inv = 1.0f / sqrtf(sh[0]);
  if (t == 0) ws[OFF_INV] = inv;
  ws[OFF_DG + t] = -ws[OFF_RSUM + t] * inv;
}

// scale WN in place; quantize delays (clip to L-1, matching JAX gather clip)
__global__ void k3_scale(const float* __restrict__ dist,
                         const float* __restrict__ theta,
                         float* ws) {
  int idx = blockIdx.x * 256 + threadIdx.x;       // < Nn*Nn
  float inv = ws[OFF_INV];
  ws[OFF_WN + idx] *= inv;
  float mu = theta[15];
  int d = (int)(dist[idx] / (1.5f + fmaxf(mu, 0.0f)));
  if (d > Ll - 1) d = Ll - 1;
  if (d < 0) d = 0;
  ((unsigned short*)(ws + OFF_DEL))[idx] = (unsigned short)d;
}

// ring init from hE, noise transpose (b,h,i)->contiguous, lm column-centering
__global__ void k4_pack(const float* __restrict__ hE,
                        const float* __restrict__ noise,
                        const float* __restrict__ lm,
                        float* ws) {
  int idx = blockIdx.x * 256 + threadIdx.x;
  if (idx < Nn * RP) {
    int j = idx >> 9, p = idx & (RP - 1);
    ws[OFF_RING + idx] = (p < Ll) ? hE[j*Ll + p] : 0.0f;
  } else if (idx < Nn * RP + TS * Nn) {
    int k = idx - Nn * RP;
    int t = k >> 9, i = k & (Nn - 1);
    int b = t / Hh, h = t - b * Hh;
    // noise_in (N,H,B,3), channel 0 only
    ws[OFF_NOISE + k] = noise[i*(Hh*Bb*3) + h*(Bb*3) + b*3 + 0];
  } else {
    int k = idx - Nn * RP - TS * Nn;                // < EEGc*Nn
    int e = k >> 9, n = k & (Nn - 1);
    float sum = 0.0f;
    for (int ee = 0; ee < EEGc; ++ee) sum += lm[ee*Nn + n];
    ws[OFF_LMT + k] = lm[e*Nn + n] - sum * (1.0f / EEGc);
  }
}

// ---------------------------------------------------------------------------
__device__ __forceinline__ void grid_barrier(unsigned* cnt, unsigned* gen) {
  __threadfence();                 // release: my stores reach device scope
  __syncthreads();
  if (threadIdx.x == 0) {
    unsigned old = __atomic_load_n(gen, __ATOMIC_RELAXED);
    if (atomicAdd(cnt, 1u) == NWG - 1) {
      __atomic_store_n(cnt, 0u, __ATOMIC_RELAXED);
      __atomic_store_n(gen, old + 1u, __ATOMIC_RELEASE);
    } else {
      while (__atomic_load_n(gen, __ATOMIC_RELAXED) == old)
        __builtin_amdgcn_s_sleep(2);
    }
  }
  __syncthreads();
  __threadfence();                 // acquire: invalidate stale L0/WGP$
}

__launch_bounds__(Nn, 1)
__global__ void k5_main(const float* __restrict__ inp,    // (H,B,2)
                        const float* __restrict__ Win,    // (N,2)
                        const float* __restrict__ hx,     // (N,6)
                        const float* __restrict__ theta,
                        float* __restrict__ ws,
                        float* __restrict__ out) {        // (B,EEG)
  const int i = threadIdx.x;                 // node owned by this thread
  const int g = blockIdx.x;                  // j-chunk owner
  unsigned* bar = (unsigned*)ws;
  const float* WN   = ws + OFF_WN;
  float*       ring = ws + OFF_RING;
  const unsigned short* del = (const unsigned short*)(ws + OFF_DEL);
  const float* noiseT = ws + OFF_NOISE;
  const float* lmT  = ws + OFF_LMT;
  float*       part = ws + OFF_PART;

  const float g_   = 0.01f + reluf(theta[0]);
  const float c1_  = 0.01f + reluf(theta[1]);
  const float c2_  = 0.01f + reluf(theta[2]);
  const float c3_  = 0.01f + reluf(theta[3]);
  const float c4_  = 0.01f + reluf(theta[4]);
  const float sin_ = 150.0f + reluf(theta[5]);
  const float A_   = reluf(theta[6]);
  const float a1   = 1.0f + reluf(theta[7]);
  const float Bp_  = reluf(theta[8]);
  const float b1   = 1.0f + reluf(theta[9]);
  const float vmax = theta[10], v0 = theta[11], r = theta[12];
  const float cy0  = theta[13], y0 = theta[14];

  float M  = hx[i*6+0], E  = hx[i*6+1], I  = hx[i*6+2];
  float Mv = hx[i*6+3], Ev = hx[i*6+4], Iv = hx[i*6+5];
  const float dgi = ws[OFF_DG + i];
  const float wi0 = Win[i*2+0], wi1 = Win[i*2+1];

  __shared__ float sEI[Nn];

  // per-lane constants for branchless WMMA B-operand construction
  const int lane = i & 31;
  const float is0  = (lane == 0)  ? 1.0f : 0.0f;
  const float is16 = (lane == 16) ? 1.0f : 0.0f;

  unsigned head = 0;
  for (int t = 0; t < TS; ++t) {
    // ---- phase 1: partial LEd over this WG's 32 ring rows (all coalesced)
    float acc = 0.0f;
    const int j0 = g * CHUNK;
    #pragma unroll 4
    for (int jj = 0; jj < CHUNK; ++jj) {
      int j = j0 + jj;
      unsigned d = del[j*Nn + i];                       // delays[j][i]
      float ev = ring[j*RP + ((head + d) & (RP - 1))];  // hE[j, delay]
      acc = fmaf(WN[j*Nn + i], ev, acc);                // w_n[i][j] (symmetric)
    }
    float* pb = part + (t & 1) * (NWG * Nn);
    pb[g*Nn + i] = acc;

    grid_barrier(bar, bar + 1);

    // ---- phase 2: redundant identical state update in every WG (lockstep)
    float LEd = 0.0f;
    #pragma unroll
    for (int gg = 0; gg < NWG; ++gg) LEd += pb[gg*Nn + i];

    int b = t / Hh, h = t - b * Hh;
    float noiseE = noiseT[t*Nn + i];
    float u0 = inp[(h*Bb + b)*2 + 0];
    float u1 = inp[(h*Bb + b)*2 + 1];
    float stim = wi0*u0 + wi1*u1;

    float rM = vmax / (1.0f + expf(-r * ((E - I) - v0)));
    float sM2 = vmax / (1.0f + expf(-r * (c1_*M - v0)));
    float sM4 = vmax / (1.0f + expf(-r * (c3_*M - v0)));
    float rE = sin_*noiseE + g_*(LEd + dgi*E) + c2_*sM2;
    float rI = c4_*sM4;

    float nM = fmaf(DTc, Mv, M);
    float nE = fmaf(DTc, Ev, E);
    float nI = fmaf(DTc, Iv, I);
    float uM = 500.0f * tanhf(rM * 0.002f);
    float uE = stim + 500.0f * tanhf(rE * 0.002f);
    float uI = 500.0f * tanhf(rI * 0.002f);
    float nMv = Mv + DTc * (A_*a1*uM  - 2.0f*a1*Mv - a1*a1*M);
    float nEv = Ev + DTc * (A_*a1*uE  - 2.0f*a1*Ev - a1*a1*E);
    float nIv = Iv + DTc * (Bp_*b1*uI - 2.0f*b1*Iv - b1*b1*I);
    M = nM; E = nE; I = nI; Mv = nMv; Ev = nEv; Iv = nIv;

    head = (head + (RP - 1)) & (RP - 1);            // head--
    if ((i >> 5) == g) ring[i*RP + head] = E;       // only owner WG writes row i

    // ---- EEG projection every 20 steps: lm_t(64x512) @ (E-I) via WMMA f32
    if (h == Hh - 1) {
      sEI[i] = E - I;
      __syncthreads();
      if (g == 0 && (i >> 5) < 4) {                 // 4 full waves, EXEC all-1s
        const int wv = i >> 5;
        const int ebase = wv * 16;
        const int m = lane & 15;
        const int koff = (lane >> 4) << 1;          // lanes 16-31 hold K+2
        const float* Arow = lmT + (ebase + m) * Nn;
        const v4f* sEI4 = (const v4f*)sEI;          // 16B-aligned broadcast
        v8f c = {0.f, 0.f, 0.f, 0.f, 0.f, 0.f, 0.f, 0.f};
        for (int k0 = 0; k0 < Nn; k0 += 4) {
          v2f av = *(const v2f*)(Arow + k0 + koff); // A tile (16x4 f32 layout)
          v4f s4 = sEI4[k0 >> 2];                   // one ds_load_b128, all lanes
          v2f bv;                                   // only column N=0 nonzero
          bv.x = fmaf(is0, s4.x, is16 * s4.z);      // branchless: EXEC untouched
          bv.y = fmaf(is0, s4.y, is16 * s4.w);
          c = __builtin_amdgcn_wmma_f32_16x16x4_f32(
                false, av, false, bv, (short)0, c, false, false);
        }
        if (lane == 0) {                            // D[M=p][N=0], M=0..7
          #pragma unroll
          for (int p = 0; p < 8; ++p)
            out[b*EEGc + ebase + p] = 0.01f*cy0*c[p] - y0;
        } else if (lane == 16) {                    // M=8..15
          #pragma unroll
          for (int p = 0; p < 8; ++p)
            out[b*EEGc + ebase + 8 + p] = 0.01f*cy0*c[p] - y0;
        }
      }
    }
    __threadfence();            // own ring writes visible to own next gathers
    __syncthreads();
  }
}

// ---------------------------------------------------------------------------
extern "C" void kernel_launch(void* const* d_in, const int* in_sizes, int n_in,
                              void* d_out, int out_size, void* d_ws, size_t ws_size,
                              hipStream_t stream) {
  (void)in_sizes; (void)n_in; (void)out_size; (void)ws_size;
  const float* inp   = (const float*)d_in[0];   // input   (H,B,2)
  const float* noise = (const float*)d_in[1];   // noise_in(N,H,B,3)
  const float* hx    = (const float*)d_in[3];   // hx      (N,6)
  const float* hE    = (const float*)d_in[4];   // hE      (N,L)
  const float* sc    = (const float*)d_in[5];   // sc      (N,N)
  const float* dist  = (const float*)d_in[6];   // dist    (N,N)
  const float* wbb   = (const float*)d_in[7];   // w_bb    (N,N)
  const float* Win   = (const float*)d_in[8];   // W_in    (N,2)
  const float* lm    = (const float*)d_in[10];  // lm      (EEG,N)
  const float* theta = (const float*)d_in[11];  // theta   (16,)
  float* ws  = (float*)d_ws;
  float* out = (float*)d_out;

  k0_init<<<1, 64, 0, stream>>>((unsigned*)d_ws);
  k1_rows<<<Nn, 256, 0, stream>>>(wbb, sc, ws);
  k2_norm<<<1, Nn, 0, stream>>>(ws);
  k3_scale<<<(Nn*Nn)/256, 256, 0, stream>>>(dist, theta, ws);
  k4_pack<<<(Nn*RP + TS*Nn + EEGc*Nn)/256, 256, 0, stream>>>(hE, noise, lm, ws);
  k5_main<<<NWG, Nn, 0, stream>>>(inp, Win, hx, theta, ws, out);
}